// GCNFlat_34110630265034
// MI455X (gfx1250) — compile-verified
//
#include <hip/hip_runtime.h>
#include <hip/hip_bf16.h>

typedef float v2f __attribute__((ext_vector_type(2)));
typedef float v8f __attribute__((ext_vector_type(8)));

#define DFEAT 128
#define NGRAPHS 128
#define NCLASSES 10

// ---------------------------------------------------------------------------
// GEMM: H[n,:] = X[n,:] @ W, X is [nrows,128], W is [128,128] row-major.
// One block = 256 threads = 8 waves. Block owns 16 rows; wave w owns the
// 16-column slab [16w,16w+16). Each wave runs 32 x v_wmma_f32_16x16x4_f32.
// A tile (16x128) staged in LDS; W read from global (64KB, stays hot in L2/WGP$).
// ---------------------------------------------------------------------------
__global__ __launch_bounds__(256) void gcn_gemm_wmma(
    const float* __restrict__ X, const float* __restrict__ W,
    float* __restrict__ H, int nrows) {
  __shared__ float sA[16 * DFEAT];  // 8 KB
  const int tid = threadIdx.x;
  const int row0 = blockIdx.x * 16;

  // Cooperative load of the 16x128 A tile (512 float4, 2 per thread), zero-pad OOB rows.
  for (int i = tid; i < 16 * (DFEAT / 4); i += 256) {
    const int r = i >> 5;          // 0..15
    const int c4 = i & 31;         // 0..31
    const int gr = row0 + r;
    float4 v = make_float4(0.f, 0.f, 0.f, 0.f);
    if (gr < nrows) v = ((const float4*)X)[(size_t)gr * (DFEAT / 4) + c4];
    ((float4*)sA)[i] = v;
  }
  __syncthreads();

  const int wave = tid >> 5;       // 0..7 -> output column tile
  const int lane = tid & 31;
  const int n0 = wave * 16;
  const int m = lane & 15;         // row within tile (A) / col within tile (B,C)
  const int hi = lane >> 4;        // half-wave selector

  v8f acc = {};                    // 16x16 f32 accumulator (8 VGPRs)
#pragma unroll
  for (int k = 0; k < DFEAT; k += 4) {
    // A 16x4 fragment: lanes 0-15 hold K={k,k+1}, lanes 16-31 hold K={k+2,k+3}
    v2f a, b;
    a[0] = sA[m * DFEAT + k + 2 * hi];
    a[1] = sA[m * DFEAT + k + 2 * hi + 1];
    // B 4x16 fragment: VGPR0 rows K={k,k+1}, VGPR1 rows K={k+2,k+3}
    b[0] = W[(size_t)(k + hi) * DFEAT + n0 + m];
    b[1] = W[(size_t)(k + 2 + hi) * DFEAT + n0 + m];
    acc = __builtin_amdgcn_wmma_f32_16x16x4_f32(
        /*neg_a=*/false, a, /*neg_b=*/false, b,
        /*c_mod=*/(short)0, acc, /*reuse_a=*/false, /*reuse_b=*/false);
  }

  // C/D layout: VGPR r -> M = r + 8*hi, N = n0 + m
  const int n = n0 + m;
#pragma unroll
  for (int r = 0; r < 8; ++r) {
    const int gm = row0 + r + hi * 8;
    if (gm < nrows) H[(size_t)gm * DFEAT + n] = acc[r];
  }
}

// ---------------------------------------------------------------------------
// Degree: deg[dst] += 1 per edge (self-loop +1 folded into rsqrt pass).
// ---------------------------------------------------------------------------
__global__ void gcn_deg(const int* __restrict__ dst, float* __restrict__ deg,
                        int nedges) {
  const int e = blockIdx.x * blockDim.x + threadIdx.x;
  if (e < nedges) atomicAdd(&deg[dst[e]], 1.0f);
}

__global__ void gcn_dinv(float* __restrict__ deg, int n) {
  const int i = blockIdx.x * blockDim.x + threadIdx.x;
  if (i < n) deg[i] = rsqrtf(deg[i] + 1.0f);
}

// ---------------------------------------------------------------------------
// Edge scatter: agg[dst,:] += h[src,:] * dinv[src]*dinv[dst].
// 32 threads per edge, float4 gather + 4 f32 atomics each. h/agg live in L2.
// ---------------------------------------------------------------------------
__global__ __launch_bounds__(256) void gcn_edge_agg(
    const int* __restrict__ src, const int* __restrict__ dst,
    const float* __restrict__ dinv, const float* __restrict__ H,
    float* __restrict__ agg, int nedges) {
  const long long t = (long long)blockIdx.x * blockDim.x + threadIdx.x;
  const long long e = t >> 5;
  if (e >= nedges) return;
  const int f4 = (int)(t & 31);    // which float4 of the 128 features
  const int s = src[e];
  const int d = dst[e];
  const float norm = dinv[s] * dinv[d];
  const float4 v = ((const float4*)(H + (size_t)s * DFEAT))[f4];
  float* out = agg + (size_t)d * DFEAT + f4 * 4;
  atomicAdd(out + 0, v.x * norm);
  atomicAdd(out + 1, v.y * norm);
  atomicAdd(out + 2, v.z * norm);
  atomicAdd(out + 3, v.w * norm);
}

// ---------------------------------------------------------------------------
// Epilogue: out = agg + dinv^2 * h + b, optional ReLU.
// ---------------------------------------------------------------------------
__global__ __launch_bounds__(256) void gcn_finalize(
    const float* __restrict__ agg, const float* __restrict__ H,
    const float* __restrict__ dinv, const float* __restrict__ bias,
    float* __restrict__ out, int n, int do_relu) {
  const long long t = (long long)blockIdx.x * blockDim.x + threadIdx.x;
  if (t >= (long long)n * DFEAT) return;
  const int node = (int)(t >> 7);
  const int d = (int)(t & (DFEAT - 1));
  const float di = dinv[node];
  float v = agg[t] + di * di * H[t] + bias[d];
  if (do_relu) v = fmaxf(v, 0.0f);
  out[t] = v;
}

// ---------------------------------------------------------------------------
// Mean-pool accumulation: pooled[batch[n],:] += x[n,:]; counts[g] += 1.
// ---------------------------------------------------------------------------
__global__ __launch_bounds__(256) void gcn_pool(
    const float* __restrict__ x, const int* __restrict__ batch,
    float* __restrict__ pooled, float* __restrict__ counts, int n) {
  const long long t = (long long)blockIdx.x * blockDim.x + threadIdx.x;
  if (t >= (long long)n * DFEAT) return;
  const int node = (int)(t >> 7);
  const int d = (int)(t & (DFEAT - 1));
  const int g = batch[node];
  atomicAdd(&pooled[(size_t)g * DFEAT + d], x[t]);
  if (d == 0) atomicAdd(&counts[g], 1.0f);
}

// ---------------------------------------------------------------------------
// Head: mean -> 128x10 linear -> stable softmax. One block (128 thr) / graph.
// ---------------------------------------------------------------------------
__global__ __launch_bounds__(128) void gcn_head(
    const float* __restrict__ pooled, const float* __restrict__ counts,
    const float* __restrict__ linW, const float* __restrict__ linb,
    float* __restrict__ out) {
  __shared__ float mvec[DFEAT];
  __shared__ float red[DFEAT];
  __shared__ float logits[NCLASSES];
  const int g = blockIdx.x;
  const int d = threadIdx.x;
  const float cnt = fmaxf(counts[g], 1.0f);
  mvec[d] = pooled[(size_t)g * DFEAT + d] / cnt;
  __syncthreads();
  for (int c = 0; c < NCLASSES; ++c) {
    red[d] = mvec[d] * linW[(size_t)d * NCLASSES + c];
    __syncthreads();
    for (int s = 64; s > 0; s >>= 1) {
      if (d < s) red[d] += red[d + s];
      __syncthreads();
    }
    if (d == 0) logits[c] = red[0] + linb[c];
    __syncthreads();
  }
  if (d == 0) {
    float mx = logits[0];
    for (int c = 1; c < NCLASSES; ++c) mx = fmaxf(mx, logits[c]);
    float e[NCLASSES], sum = 0.f;
    for (int c = 0; c < NCLASSES; ++c) { e[c] = __expf(logits[c] - mx); sum += e[c]; }
    const float inv = 1.0f / sum;
    for (int c = 0; c < NCLASSES; ++c) out[(size_t)g * NCLASSES + c] = e[c] * inv;
  }
}

// ---------------------------------------------------------------------------
extern "C" void kernel_launch(void* const* d_in, const int* in_sizes, int n_in,
                              void* d_out, int out_size, void* d_ws, size_t ws_size,
                              hipStream_t stream) {
  // setup_inputs order: x, edge_index, batch, W0, b0, W1, b1, W2, b2, lin_W, lin_b
  const float* x      = (const float*)d_in[0];
  const int*   eidx   = (const int*)d_in[1];
  const int*   batch  = (const int*)d_in[2];
  const float* W[3]   = {(const float*)d_in[3], (const float*)d_in[5], (const float*)d_in[7]};
  const float* B[3]   = {(const float*)d_in[4], (const float*)d_in[6], (const float*)d_in[8]};
  const float* linW   = (const float*)d_in[9];
  const float* linb   = (const float*)d_in[10];
  float* out = (float*)d_out;

  const int N = in_sizes[0] / DFEAT;     // 100000
  const int E = in_sizes[1] / 2;         // 1600000
  const int* src = eidx;
  const int* dst = eidx + E;

  // Workspace carve-up (floats)
  float* xA     = (float*)d_ws;                     // N*128  (layer activations)
  float* hB     = xA + (size_t)N * DFEAT;           // N*128  (GEMM output)
  float* aggC   = hB + (size_t)N * DFEAT;           // N*128  (edge aggregation)
  float* dinv   = aggC + (size_t)N * DFEAT;         // N      (deg -> dinv in place)
  float* pooled = dinv + N;                         // 128*128
  float* counts = pooled + (size_t)NGRAPHS * DFEAT; // 128

  const int nf = N * DFEAT;

  // Degree + dinv
  hipMemsetAsync(dinv, 0, (size_t)N * sizeof(float), stream);
  gcn_deg<<<(E + 255) / 256, 256, 0, stream>>>(dst, dinv, E);
  gcn_dinv<<<(N + 255) / 256, 256, 0, stream>>>(dinv, N);

  const float* layer_in = x;
  for (int l = 0; l < 3; ++l) {
    gcn_gemm_wmma<<<(N + 15) / 16, 256, 0, stream>>>(layer_in, W[l], hB, N);
    hipMemsetAsync(aggC, 0, (size_t)nf * sizeof(float), stream);
    {
      const long long threads = (long long)E * 32;
      const int blocks = (int)((threads + 255) / 256);
      gcn_edge_agg<<<blocks, 256, 0, stream>>>(src, dst, dinv, hB, aggC, E);
    }
    gcn_finalize<<<(nf + 255) / 256, 256, 0, stream>>>(aggC, hB, dinv, B[l], xA, N,
                                                       /*do_relu=*/(l < 2) ? 1 : 0);
    layer_in = xA;
  }

  // Global mean pool + head
  hipMemsetAsync(pooled, 0, (size_t)NGRAPHS * DFEAT * sizeof(float), stream);
  hipMemsetAsync(counts, 0, (size_t)NGRAPHS * sizeof(float), stream);
  gcn_pool<<<(nf + 255) / 256, 256, 0, stream>>>(xA, batch, pooled, counts, N);
  gcn_head<<<NGRAPHS, DFEAT, 0, stream>>>(pooled, counts, linW, linb, out);
}